// EquivariantProjectorviaSchur_66838281060680
// MI455X (gfx1250) — compile-verified
//
#include <hip/hip_runtime.h>

// ---------------------------------------------------------------------------
// EquivariantProjectorviaSchur — MI455X (gfx1250) fp32 WMMA implementation
//
// Per (c_out, c_in) 64x64 block:   Wp = Uy^T * Wb * Ux
//                                  Wp *= mask ; 2x2 Schur symmetrize (hard)
//                                  out = Uy * Wp * Ux^T
// 4 x (64x64x64) matmuls per block, all done with v_wmma_f32_16x16x4_f32.
// ---------------------------------------------------------------------------

#define O_DIM 64
#define C_OUT 96
#define C_IN  96
#define NDIM  6144          // C_OUT*O_DIM == C_IN*O_DIM
#define LS    68            // padded LDS row stride (floats): 68%64==4 -> no
                            // bank conflicts on column walks, 16B aligned rows

typedef __attribute__((ext_vector_type(2))) float v2f;
typedef __attribute__((ext_vector_type(8))) float v8f;

__device__ __forceinline__ v8f wmma_f32(v2f a, v2f b, v8f c) {
  // D = A(16x4) * B(4x16) + C(16x16), fp32. 8-arg form per probe:
  // (neg_a, A, neg_b, B, c_mod, C, reuse_a, reuse_b)
  return __builtin_amdgcn_wmma_f32_16x16x4_f32(
      false, a, false, b, (short)0, c, false, false);
}

// A fragment: 16x4 (MxK). Lanes 0-15 -> M=0..15 (K = kb*4 + {0,1}),
// lanes 16-31 -> M=0..15 (K = kb*4 + {2,3}).  TA=true reads A^T from LDS.
template <bool TA>
__device__ __forceinline__ v2f load_a(const float* A, int mt, int kb, int lane) {
  int m = mt * 16 + (lane & 15);
  int k = kb * 4 + ((lane >> 4) << 1);
  v2f r;
  if (!TA) { r.x = A[m * LS + k];     r.y = A[m * LS + k + 1];   }
  else     { r.x = A[k * LS + m];     r.y = A[(k + 1) * LS + m]; }
  return r;
}

// B fragment: 4x16 (KxN). Lanes 0-15 -> N=0..15 (K = kb*4 + {0,1}),
// lanes 16-31 -> N=0..15 (K = kb*4 + {2,3}).  TB=true reads B^T from LDS.
template <bool TB>
__device__ __forceinline__ v2f load_b(const float* B, int nt, int kb, int lane) {
  int n = nt * 16 + (lane & 15);
  int k = kb * 4 + ((lane >> 4) << 1);
  v2f r;
  if (!TB) { r.x = B[k * LS + n];     r.y = B[(k + 1) * LS + n]; }
  else     { r.x = B[n * LS + k];     r.y = B[n * LS + k + 1];   }
  return r;
}

// C/D layout: VGPR v -> row (v + 8*(lane>>4)), col = lane&15 within the tile.
__device__ __forceinline__ void store_tile(float* Out, v8f acc, int mt, int nt,
                                           int lane) {
  int col   = nt * 16 + (lane & 15);
  int rbase = mt * 16 + ((lane >> 4) << 3);
#pragma unroll
  for (int v = 0; v < 8; ++v) Out[(rbase + v) * LS + col] = acc[v];
}

// One 64x64x64 matmul stage: each wave computes 2 of the 16 output tiles.
template <bool TA, bool TB>
__device__ __forceinline__ void stage_mm(const float* A, const float* B,
                                         float* Out, int w, int lane) {
  int mt = w >> 1, nt0 = (w & 1) * 2;
  v8f acc0 = {};
  v8f acc1 = {};
#pragma unroll
  for (int kb = 0; kb < 16; ++kb) {
    v2f a  = load_a<TA>(A, mt, kb, lane);
    v2f b0 = load_b<TB>(B, nt0,     kb, lane);
    v2f b1 = load_b<TB>(B, nt0 + 1, kb, lane);
    acc0 = wmma_f32(a, b0, acc0);
    acc1 = wmma_f32(a, b1, acc1);
  }
  store_tile(Out, acc0, mt, nt0,     lane);
  store_tile(Out, acc1, mt, nt0 + 1, lane);
}

__global__ __launch_bounds__(256) void schur_proj_kernel(
    const float* __restrict__ W, const float* __restrict__ Uy,
    const float* __restrict__ Ux, const float* __restrict__ mask,
    const int* __restrict__ brows, const int* __restrict__ bcols,
    float* __restrict__ out, int nb) {
  __shared__ float Wl[O_DIM * LS];   // W block / Wp
  __shared__ float Tl[O_DIM * LS];   // intermediates T1 / T3
  __shared__ float Uyl[O_DIM * LS];
  __shared__ float Uxl[O_DIM * LS];

  const int tid  = threadIdx.x;
  const int lane = tid & 31;
  const int w    = tid >> 5;
  const int cOut = blockIdx.x / C_IN;
  const int cIn  = blockIdx.x % C_IN;

  const float* Wblk = W + (size_t)cOut * O_DIM * NDIM + (size_t)cIn * O_DIM;

  // Cooperative loads: 64x64 fp32 = 1024 float4 each, 4 float4 per thread.
#pragma unroll
  for (int k = 0; k < 4; ++k) {
    int f  = tid + k * 256;       // float4 index 0..1023
    int p  = f >> 4;              // row 0..63
    int q4 = (f & 15) << 2;       // col (multiple of 4)
    *(float4*)&Wl[p * LS + q4]  = *(const float4*)&Wblk[(size_t)p * NDIM + q4];
    *(float4*)&Uyl[p * LS + q4] = *(const float4*)&Uy[p * O_DIM + q4];
    *(float4*)&Uxl[p * LS + q4] = *(const float4*)&Ux[p * O_DIM + q4];
  }
  __syncthreads();

  // T1 = Uy^T * W
  stage_mm<true, false>(Uyl, Wl, Tl, w, lane);
  __syncthreads();

  // Wp = T1 * Ux   (overwrite Wl)
  stage_mm<false, false>(Tl, Uxl, Wl, w, lane);
  __syncthreads();

  // Elementwise mask (mask streamed from L2-resident global).
#pragma unroll
  for (int k = 0; k < 16; ++k) {
    int e = tid + k * 256;
    Wl[(e >> 6) * LS + (e & 63)] *= mask[e];
  }
  __syncthreads();

  // Hard Schur symmetrization of the nb matched 2x2 blocks (SOFTNESS == 0).
  if (tid < nb) {
    int r0 = brows[4 * tid],     c0 = bcols[4 * tid];
    int c1 = bcols[4 * tid + 1], r1 = brows[4 * tid + 2];
    float g00 = Wl[r0 * LS + c0], g01 = Wl[r0 * LS + c1];
    float g10 = Wl[r1 * LS + c0], g11 = Wl[r1 * LS + c1];
    float alpha = 0.5f * (g00 + g11);
    float beta  = 0.5f * (g01 - g10);
    Wl[r0 * LS + c0] = alpha;  Wl[r0 * LS + c1] = beta;
    Wl[r1 * LS + c0] = -beta;  Wl[r1 * LS + c1] = alpha;
  }
  __syncthreads();

  // T3 = Uy * Wp
  stage_mm<false, false>(Uyl, Wl, Tl, w, lane);
  __syncthreads();

  // out = T3 * Ux^T  -> straight to global (coalesced per half-wave row)
  {
    int mt = w >> 1, nt0 = (w & 1) * 2;
    v8f acc0 = {};
    v8f acc1 = {};
#pragma unroll
    for (int kb = 0; kb < 16; ++kb) {
      v2f a  = load_a<false>(Tl, mt, kb, lane);
      v2f b0 = load_b<true>(Uxl, nt0,     kb, lane);
      v2f b1 = load_b<true>(Uxl, nt0 + 1, kb, lane);
      acc0 = wmma_f32(a, b0, acc0);
      acc1 = wmma_f32(a, b1, acc1);
    }
    size_t rbase = (size_t)cOut * O_DIM + mt * 16 + ((lane >> 4) << 3);
    size_t cb0   = (size_t)cIn * O_DIM + nt0 * 16 + (lane & 15);
#pragma unroll
    for (int v = 0; v < 8; ++v) {
      out[(rbase + v) * NDIM + cb0]      = acc0[v];
      out[(rbase + v) * NDIM + cb0 + 16] = acc1[v];
    }
  }
}

extern "C" void kernel_launch(void* const* d_in, const int* in_sizes, int n_in,
                              void* d_out, int out_size, void* d_ws,
                              size_t ws_size, hipStream_t stream) {
  const float* W    = (const float*)d_in[0];
  const float* Uy   = (const float*)d_in[1];
  const float* Ux   = (const float*)d_in[2];
  const float* mask = (const float*)d_in[3];
  const int*   br   = (const int*)d_in[4];
  const int*   bc   = (const int*)d_in[5];
  float*       out  = (float*)d_out;
  int nb = in_sizes[4] / 4;   // number of matched 2x2 blocks (72)

  dim3 grid(C_OUT * C_IN);    // 9216 independent 64x64 blocks
  dim3 block(256);            // 8 waves of 32
  schur_proj_kernel<<<grid, block, 0, stream>>>(W, Uy, Ux, mask, br, bc, out, nb);
}